// HardSigmoidRT_41085657153639
// MI455X (gfx1250) — compile-verified
//
#include <hip/hip_runtime.h>
#include <math.h>

#define N_IMG 128
#define H_DIM 1024
#define W_DIM 512
#define HW (H_DIM * W_DIM)     // 524288
#define HID 64

typedef float v2f __attribute__((ext_vector_type(2)));
typedef float v4f __attribute__((ext_vector_type(4)));
typedef float v8f __attribute__((ext_vector_type(8)));

// ---------------------------------------------------------------------------
// Kernel 1: eta MLP.  One workgroup, 8 waves.  Uses V_WMMA_F32_16X16X4_F32
// (K=4 f32 WMMA exactly matches the 4-wide feature dimension).
// Wave w owns M-tile w (rows 16w..16w+15 of the 128-row batch).
// ---------------------------------------------------------------------------
__global__ __launch_bounds__(256) void eta_mlp_kernel(
    const float* __restrict__ rt_, const float* __restrict__ noise,
    const float* __restrict__ Xmin, const float* __restrict__ Xmax,
    const float* __restrict__ Ymin, const float* __restrict__ Ymax,
    const float* __restrict__ W1, const float* __restrict__ b1,
    const float* __restrict__ W2, const float* __restrict__ b2,
    float* __restrict__ eta_out)
{
    __shared__ float s_xn[N_IMG][4];     // normalized features
    __shared__ float s_h[N_IMG][HID];    // hidden layer (post-ReLU)
    __shared__ float s_W1[4][HID];       // [4][64] row-major
    __shared__ float s_W2[HID][4];       // [64][4] row-major

    const int tid = threadIdx.x;

    // Stage weights (each has exactly 256 elements).
    s_W1[tid >> 6][tid & 63] = W1[tid];
    s_W2[tid >> 2][tid & 3]  = W2[tid];

    // Threads 0..127 compute one feature row each.
    if (tid < N_IMG) {
        const float r0 = 1.0f / (1.0f + expf(-rt_[0]));
        const float r1 = 1.0f / (1.0f + expf(-rt_[1]));
        const float r2 = 1.0f / (1.0f + expf(-rt_[2]));
        const float RT0 = r0 * (Xmax[0] - Xmin[0]) + Xmin[0];
        const float RT1 = r1 * (Xmax[1] - Xmin[1]) + Xmin[1];
        const float RT2 = r2 * (Xmax[2] - Xmin[2]) + Xmin[2];
        const float n0 = noise[tid * 4 + 0];
        const float n1 = noise[tid * 4 + 1];
        const float n2 = noise[tid * 4 + 2];
        const float e0 = RT0 * n0;
        const float e1 = RT1 * n1;
        const float e2 = RT2 * n2;
        const float e3 = e1 / e2;
        s_xn[tid][0] = (e0 - Xmin[0]) / (Xmax[0] - Xmin[0]);
        s_xn[tid][1] = (e1 - Xmin[1]) / (Xmax[1] - Xmin[1]);
        s_xn[tid][2] = (e2 - Xmin[2]) / (Xmax[2] - Xmin[2]);
        s_xn[tid][3] = (e3 - Xmin[3]) / (Xmax[3] - Xmin[3]);
    }
    __syncthreads();

    const int wave  = tid >> 5;          // 0..7  -> M tile
    const int lane  = tid & 31;
    const int l15   = lane & 15;
    const bool hi   = lane >= 16;
    const int mbase = wave * 16;

    // ---- Layer 1: xn[128,4] @ W1[4,64] -> h, one K=4 WMMA per 16x16 tile ---
    // A fragment (16x4 f32): lanes0-15 K=0,1 ; lanes16-31 K=2,3
    v2f a1;
    a1.x = s_xn[mbase + l15][hi ? 2 : 0];
    a1.y = s_xn[mbase + l15][hi ? 3 : 1];

    #pragma unroll
    for (int nt = 0; nt < 4; ++nt) {
        // B fragment (4x16 f32): v0 = K row 0 / 2, v1 = K row 1 / 3
        v2f bfrag;
        bfrag.x = s_W1[hi ? 2 : 0][nt * 16 + l15];
        bfrag.y = s_W1[hi ? 3 : 1][nt * 16 + l15];
        v8f c = {};
        c = __builtin_amdgcn_wmma_f32_16x16x4_f32(
                false, a1, false, bfrag, (short)0, c, false, false);
        // bias + ReLU, scatter C tile (VGPR v -> row v / v+8) to LDS
        const float bias = b1[nt * 16 + l15];
        #pragma unroll
        for (int v = 0; v < 8; ++v) {
            const int m = v + (hi ? 8 : 0);
            const float val = c[v] + bias;
            s_h[mbase + m][nt * 16 + l15] = val > 0.0f ? val : 0.0f;
        }
    }
    __syncthreads();

    // ---- Layer 2: h[128,64] @ W2[64,4] -> 16 accumulating K=4 WMMAs --------
    v8f acc = {};
    #pragma unroll
    for (int kc = 0; kc < 16; ++kc) {
        const int k0 = kc * 4;
        v2f a2, b2f;
        a2.x = s_h[mbase + l15][k0 + (hi ? 2 : 0)];
        a2.y = s_h[mbase + l15][k0 + (hi ? 3 : 1)];
        // B is 64x16 with only N=0..3 valid (zero-padded); ternary keeps EXEC full
        b2f.x = (l15 < 4) ? s_W2[k0 + (hi ? 2 : 0)][l15] : 0.0f;
        b2f.y = (l15 < 4) ? s_W2[k0 + (hi ? 3 : 1)][l15] : 0.0f;
        acc = __builtin_amdgcn_wmma_f32_16x16x4_f32(
                  false, a2, false, b2f, (short)0, acc, false, false);
    }

    // ---- sigmoid + denormalize, write eta[128][4] --------------------------
    const int col = l15;
    if (col < 4) {
        const float ymin = Ymin[col];
        const float yrng = Ymax[col] - ymin;
        const float bias2 = b2[col];
        #pragma unroll
        for (int v = 0; v < 8; ++v) {
            const int row = mbase + v + (hi ? 8 : 0);
            const float x = acc[v] + bias2;
            const float s = 1.0f / (1.0f + expf(-x));
            eta_out[row * 4 + col] = s * yrng + ymin;
        }
    }
}

// ---------------------------------------------------------------------------
// Kernel 2: bandwidth-bound piecewise-linear map over z (256MB in / 256MB out).
// float4 (b128) non-temporal loads/stores; eta read uniformly per block.
// ---------------------------------------------------------------------------
__global__ __launch_bounds__(256) void piecewise_kernel(
    const float* __restrict__ z, const float* __restrict__ eta,
    float* __restrict__ out)
{
    const int n = blockIdx.y;
    const float e0 = eta[n * 4 + 0];
    const float e1 = eta[n * 4 + 1];
    const float e2 = eta[n * 4 + 2];
    const float e3 = eta[n * 4 + 3];
    const float slope = (e1 - e0) / (e3 - e2);

    const size_t idx4 = (size_t)n * (HW / 4)
                      + (size_t)blockIdx.x * blockDim.x + threadIdx.x;

    const v4f* zp = (const v4f*)z;
    v4f*       op = (v4f*)out;

    v4f zv = __builtin_nontemporal_load(zp + idx4);
    v4f r;
    #pragma unroll
    for (int i = 0; i < 4; ++i) {
        const float zi  = zv[i];
        const float lin = fmaf(zi - e2, slope, e0);
        r[i] = zi < e2 ? e0 : (zi <= e3 ? lin : e1);
    }
    __builtin_nontemporal_store(r, op + idx4);
}

// ---------------------------------------------------------------------------
extern "C" void kernel_launch(void* const* d_in, const int* in_sizes, int n_in,
                              void* d_out, int out_size, void* d_ws, size_t ws_size,
                              hipStream_t stream) {
    const float* z     = (const float*)d_in[0];
    const float* rt_   = (const float*)d_in[1];
    const float* noise = (const float*)d_in[2];
    const float* Xmin  = (const float*)d_in[3];
    const float* Xmax  = (const float*)d_in[4];
    const float* Ymin  = (const float*)d_in[5];
    const float* Ymax  = (const float*)d_in[6];
    const float* W1    = (const float*)d_in[7];
    const float* b1    = (const float*)d_in[8];
    const float* W2    = (const float*)d_in[9];
    const float* b2    = (const float*)d_in[10];
    float* out = (float*)d_out;
    float* eta = (float*)d_ws;               // 128*4 floats = 2 KB scratch

    eta_mlp_kernel<<<dim3(1), dim3(256), 0, stream>>>(
        rt_, noise, Xmin, Xmax, Ymin, Ymax, W1, b1, W2, b2, eta);

    dim3 grid(HW / (4 * 256), N_IMG);        // (512, 128)
    piecewise_kernel<<<grid, dim3(256), 0, stream>>>(z, eta, out);
}